// CLALoss_86406152061536
// MI455X (gfx1250) — compile-verified
//
#include <hip/hip_runtime.h>
#include <math.h>

typedef __attribute__((ext_vector_type(2))) float v2f;
typedef __attribute__((ext_vector_type(8))) float v8f;

#define NB     16384          // batch B
#define ND     64             // embedding dim D
#define NK     256            // codes K
#define BKELEM (NB * NK)      // 4194304
#define EMELEM (NB * ND)      // 1048576

__device__ __forceinline__ float wred_sum(float v) {
#pragma unroll
    for (int m = 16; m >= 1; m >>= 1) v += __shfl_xor(v, m, 32);
    return v;
}
__device__ __forceinline__ float wred_max(float v) {
#pragma unroll
    for (int m = 16; m >= 1; m >>= 1) v = fmaxf(v, __shfl_xor(v, m, 32));
    return v;
}

// ---------------- weight row-normalize: 512 rows of 64 ----------------
__global__ void k_norm_weights(const float* __restrict__ wf_in,
                               const float* __restrict__ wi_in,
                               float* __restrict__ wf_out,
                               float* __restrict__ wi_out) {
    const int lane = threadIdx.x & 31, wave = threadIdx.x >> 5;
    const int row = blockIdx.x * 8 + wave;            // 0..511
    const float* src; float* dst; int rr;
    if (row < NK) { src = wf_in; dst = wf_out; rr = row; }
    else          { src = wi_in; dst = wi_out; rr = row - NK; }
    v2f v = *(const v2f*)(src + (size_t)rr * ND + lane * 2);
    float ss = wred_sum(v.x * v.x + v.y * v.y);
    float inv = 1.0f / fmaxf(sqrtf(ss), 1e-12f);
    v2f o; o.x = v.x * inv; o.y = v.y * inv;
    *(v2f*)(dst + (size_t)rr * ND + lane * 2) = o;
}

// ---------------- gather + row-normalize embeddings ----------------
__global__ void k_gather_norm(const float* __restrict__ ue,
                              const float* __restrict__ ide,
                              const float* __restrict__ ime,
                              const float* __restrict__ txe,
                              const int* __restrict__ users,
                              const int* __restrict__ items,
                              float* __restrict__ out_base) {
    const int lane = threadIdx.x & 31, wave = threadIdx.x >> 5;
    const int row = blockIdx.x * 8 + wave;            // 0..16383
    const float* src; const int* idxarr;
    switch (blockIdx.y) {
        case 0:  src = ue;  idxarr = users; break;
        case 1:  src = ide; idxarr = items; break;
        case 2:  src = ime; idxarr = items; break;
        default: src = txe; idxarr = items; break;
    }
    float* dst = out_base + (size_t)blockIdx.y * EMELEM;
    const int idx = idxarr[row];
    v2f v = *(const v2f*)(src + (size_t)idx * ND + lane * 2);
    float ss = wred_sum(v.x * v.x + v.y * v.y);
    float inv = 1.0f / fmaxf(sqrtf(ss), 1e-12f);
    v2f o; o.x = v.x * inv; o.y = v.y * inv;
    *(v2f*)(dst + (size_t)row * ND + lane * 2) = o;
}

// ---------------- WMMA fp32 GEMM: code[g] = X[g] @ W[g]^T ----------------
// grid.x = B/16 (M tile), grid.y = 5 (which gemm). Block = 256 (8 waves),
// each wave owns two 16x16 N-tiles; K-reduction = 16 steps of 16x16x4 fp32 WMMA.
__global__ void k_wmma_gemm(const float* __restrict__ emb_base,
                            const float* __restrict__ wfeat,
                            const float* __restrict__ wii,
                            float* __restrict__ codes) {
    const int g = blockIdx.y;
    const float* X; const float* W;
    switch (g) {
        case 0:  X = emb_base + 0 * (size_t)EMELEM; W = wfeat; break; // user @ feat
        case 1:  X = emb_base + 1 * (size_t)EMELEM; W = wfeat; break; // id   @ feat
        case 2:  X = emb_base + 1 * (size_t)EMELEM; W = wii;   break; // id   @ ii
        case 3:  X = emb_base + 2 * (size_t)EMELEM; W = wii;   break; // image@ ii
        default: X = emb_base + 3 * (size_t)EMELEM; W = wii;   break; // text @ ii
    }
    float* OUT = codes + (size_t)g * BKELEM;

    const int lane = threadIdx.x & 31;
    const int wave = threadIdx.x >> 5;
    const int l16  = lane & 15;
    const int half = lane >> 4;
    const int m0   = blockIdx.x * 16;
    const int n0a  = wave * 32;
    const int n0b  = n0a + 16;

    const float* xrow = X + (size_t)(m0 + l16) * ND + 2 * half;
    const float* wca  = W + (size_t)(n0a + l16) * ND + 2 * half;
    const float* wcb  = W + (size_t)(n0b + l16) * ND + 2 * half;

    v8f acc0 = {0.f, 0.f, 0.f, 0.f, 0.f, 0.f, 0.f, 0.f};
    v8f acc1 = {0.f, 0.f, 0.f, 0.f, 0.f, 0.f, 0.f, 0.f};
#pragma unroll
    for (int d0 = 0; d0 < ND; d0 += 4) {
        v2f a  = *(const v2f*)(xrow + d0);
        v2f b0 = *(const v2f*)(wca + d0);
        v2f b1 = *(const v2f*)(wcb + d0);
        acc0 = __builtin_amdgcn_wmma_f32_16x16x4_f32(false, a, false, b0,
                                                     (short)0, acc0, false, false);
        acc1 = __builtin_amdgcn_wmma_f32_16x16x4_f32(false, a, false, b1,
                                                     (short)0, acc1, false, false);
    }
#pragma unroll
    for (int v = 0; v < 8; ++v) {
        const int row = m0 + v + 8 * half;
        OUT[(size_t)row * NK + n0a + l16] = acc0[v];
        OUT[(size_t)row * NK + n0b + l16] = acc1[v];
    }
}

// ---------------- Sinkhorn r-step: r[k] = 1/(K * sum_b E[k,b]*c[b]) ----------------
// grid.x = K/32, grid.y = 5; block 256: 32 k's x 8 b-strides. Deterministic.
__global__ void k_sink_r(const float* __restrict__ codes,
                         const float* __restrict__ cvec,   // null => c == 1
                         float* __restrict__ rvec) {
    const int g  = blockIdx.y;
    const float* code = codes + (size_t)g * BKELEM;
    const float* cg = cvec ? cvec + (size_t)g * NB : nullptr;
    const int kk = threadIdx.x & 31, bb = threadIdx.x >> 5;
    const int k  = blockIdx.x * 32 + kk;
    float acc = 0.f;
    for (int b = bb; b < NB; b += 8) {
        float e = expf(20.f * code[(size_t)b * NK + k]);
        acc += cg ? e * cg[b] : e;
    }
    __shared__ float lds[256];
    lds[threadIdx.x] = acc;
    __syncthreads();
    if (bb == 0) {
        float s = 0.f;
        for (int j = 0; j < 8; ++j) s += lds[j * 32 + kk];   // fixed order
        rvec[g * NK + k] = 1.0f / (256.0f * s);
    }
}

// ---------------- Sinkhorn c-step: c[b] = 1/(B * sum_k E[k,b]*r[k]) ----------------
__global__ void k_sink_c(const float* __restrict__ codes,
                         const float* __restrict__ rvec,
                         float* __restrict__ cvec) {
    const int g = blockIdx.y;
    const float* cr = codes + (size_t)g * BKELEM;
    const float* rg = rvec + g * NK;
    const int lane = threadIdx.x & 31, wave = threadIdx.x >> 5;
    const int row = blockIdx.x * 8 + wave;
    const float* rowp = cr + (size_t)row * NK;
    float acc = 0.f;
#pragma unroll
    for (int j = 0; j < 8; ++j) {
        const int k = lane + 32 * j;
        acc += expf(20.f * rowp[k]) * rg[k];
    }
    float T = wred_sum(acc);
    if (lane == 0) cvec[g * NB + row] = 1.0f / (16384.0f * T);
}

// ---------------- fused per-row CE over all 8 pairs ----------------
__global__ void k_loss_rows(const float* __restrict__ codes,
                            const float* __restrict__ rvec,
                            const float* __restrict__ cvec,
                            const float* __restrict__ gamma_p,
                            float* __restrict__ partials) {
    const int lane = threadIdx.x & 31, wave = threadIdx.x >> 5;
    const int row = blockIdx.x * 8 + wave;
    const float g = fminf(fmaxf(gamma_p[0], 0.01f), 0.99f);
    const float invg = 1.0f / g;

    float cv[5][8];    // raw code values
    float eqr[5][8];   // exp(20*code) * r[k]
    float qs[5], mx[5], lse[5];
#pragma unroll
    for (int m = 0; m < 5; ++m) {
        const float* cr = codes + (size_t)m * BKELEM + (size_t)row * NK;
        const float* rm = rvec + m * NK;
        float q = 0.f, mm = -1e30f;
#pragma unroll
        for (int j = 0; j < 8; ++j) {
            const int k = lane + 32 * j;
            const float x = cr[k];
            cv[m][j] = x;
            const float e = expf(20.f * x) * rm[k];
            eqr[m][j] = e;
            q += e;
            mm = fmaxf(mm, x);
        }
        qs[m] = wred_sum(q);
        mx[m] = wred_max(mm);
        float se = 0.f;
#pragma unroll
        for (int j = 0; j < 8; ++j) se += expf((cv[m][j] - mx[m]) * invg);
        lse[m] = mx[m] * invg + logf(wred_sum(se));
    }

    // pairs: (q_user,code_id),(q_id,code_user) w=1/2;
    //        (2,3)(2,4)(3,2)(3,4)(4,2)(4,3)     w=1/6
    const int pq[8] = {0, 1, 2, 2, 3, 3, 4, 4};
    const int pc[8] = {1, 0, 3, 4, 2, 4, 2, 3};
    float dot[8];
#pragma unroll
    for (int p = 0; p < 8; ++p) {
        float d = 0.f;
#pragma unroll
        for (int j = 0; j < 8; ++j) d += eqr[pq[p]][j] * cv[pc[p]][j];
        dot[p] = wred_sum(d) * invg;   // sum_k eq * (code/g)
    }

    __shared__ float lds[8];
    if (lane == 0) {
        float Qfac[5];
#pragma unroll
        for (int m = 0; m < 5; ++m) Qfac[m] = 16384.0f * cvec[m * NB + row];
        float t[8];
#pragma unroll
        for (int p = 0; p < 8; ++p)
            t[p] = Qfac[pq[p]] * (dot[p] - lse[pc[p]] * qs[pq[p]]);
        const float contrib =
            -((t[0] + t[1]) * 0.5f +
              (t[2] + t[3] + t[4] + t[5] + t[6] + t[7]) * (1.0f / 6.0f)) *
            (1.0f / 16384.0f);
        lds[wave] = contrib;
    }
    __syncthreads();
    if (threadIdx.x == 0) {
        float s = 0.f;
        for (int w = 0; w < 8; ++w) s += lds[w];   // fixed order
        partials[blockIdx.x] = s;
    }
}

__global__ void k_finalize(const float* __restrict__ partials, float* __restrict__ out) {
    if (threadIdx.x == 0 && blockIdx.x == 0) {
        float s = 0.f;
        for (int i = 0; i < 2048; ++i) s += partials[i];   // fixed order
        out[0] = s;
    }
}

extern "C" void kernel_launch(void* const* d_in, const int* in_sizes, int n_in,
                              void* d_out, int out_size, void* d_ws, size_t ws_size,
                              hipStream_t stream) {
    const float* user_emb = (const float*)d_in[0];
    const float* id_emb   = (const float*)d_in[1];
    const float* im_emb   = (const float*)d_in[2];
    const float* tx_emb   = (const float*)d_in[3];
    const int*   users    = (const int*)d_in[4];
    const int*   items    = (const int*)d_in[5];
    const float* feat_w   = (const float*)d_in[6];
    const float* ii_w     = (const float*)d_in[7];
    const float* gamma    = (const float*)d_in[8];
    float* out = (float*)d_out;

    // workspace layout (floats)
    float* ws = (float*)d_ws;
    float* wfeat    = ws;                        // 256*64
    float* wii      = wfeat + NK * ND;           // 256*64
    float* emb      = wii + NK * ND;             // 4 * 16384*64 (user,id,image,text)
    float* codes    = emb + 4 * (size_t)EMELEM;  // 5 * 16384*256
    float* rvec     = codes + 5 * (size_t)BKELEM;// 5 * 256
    float* cvec     = rvec + 5 * NK;             // 5 * 16384
    float* partials = cvec + 5 * NB;             // 2048

    // 1) weight row normalization (512 rows)
    k_norm_weights<<<dim3(64), dim3(256), 0, stream>>>(feat_w, ii_w, wfeat, wii);

    // 2) gather + normalize 4 embedding sets
    k_gather_norm<<<dim3(NB / 8, 4), dim3(256), 0, stream>>>(
        user_emb, id_emb, im_emb, tx_emb, users, items, emb);

    // 3) five GEMMs via fp32 WMMA
    k_wmma_gemm<<<dim3(NB / 16, 5), dim3(256), 0, stream>>>(emb, wfeat, wii, codes);

    // 4) Sinkhorn: 3 iterations of (r-step, c-step); first r-step uses c==1
    for (int it = 0; it < 3; ++it) {
        k_sink_r<<<dim3(NK / 32, 5), dim3(256), 0, stream>>>(
            codes, it == 0 ? (const float*)nullptr : cvec, rvec);
        k_sink_c<<<dim3(NB / 8, 5), dim3(256), 0, stream>>>(codes, rvec, cvec);
    }

    // 5) fused CE over all 8 pairs -> per-block partials -> scalar
    k_loss_rows<<<dim3(NB / 8), dim3(256), 0, stream>>>(codes, rvec, cvec, gamma, partials);
    k_finalize<<<dim3(1), dim3(1), 0, stream>>>(partials, out);

    (void)in_sizes; (void)n_in; (void)out_size; (void)ws_size;
}